// RoipoolingLayer_56203942035848
// MI455X (gfx1250) — compile-verified
//
#include <hip/hip_runtime.h>
#include <hip/hip_bf16.h>
#include <stdint.h>

#define HFEAT 37
#define WFEAT 50
#define CCH   512
#define POOLD 7
#define SAMP  14
#define NCELL (POOLD * POOLD)

// ---- CDNA5 async global->LDS helpers -------------------------------------
// Probe-verified: __builtin_amdgcn_global_load_async_to_lds_b128 takes
// (AS1 v4i* gsrc, AS3 v4i* lds, imm offset, imm cpol).
typedef int v4i __attribute__((vector_size(16)));
typedef float f2 __attribute__((ext_vector_type(2)));
typedef __attribute__((address_space(1))) v4i gv4i_t;
typedef __attribute__((address_space(3))) v4i lv4i_t;

__device__ __forceinline__ void async_copy_b128_to_lds(const float* gsrc, float* lds_dst) {
#if defined(__gfx1250__) && __has_builtin(__builtin_amdgcn_global_load_async_to_lds_b128)
    __builtin_amdgcn_global_load_async_to_lds_b128(
        (gv4i_t*)(uintptr_t)gsrc,
        (lv4i_t*)(unsigned)(uintptr_t)lds_dst,
        0, 0);
#else
    unsigned lds_addr = (unsigned)(uintptr_t)lds_dst;
    asm volatile("global_load_async_to_lds_b128 %0, %1, off"
                 :: "v"(lds_addr), "v"(gsrc)
                 : "memory");
#endif
}

// Async loads complete in-order (ISA 08 §4.1): waiting ASYNCcnt<=8 after
// issuing a second batch of 8 guarantees the first batch of 8 has landed.
__device__ __forceinline__ void wait_async_le8() {
#if defined(__gfx1250__) && __has_builtin(__builtin_amdgcn_s_wait_asynccnt)
    __builtin_amdgcn_s_wait_asynccnt(8);
#else
    asm volatile("s_wait_asynccnt 8" ::: "memory");
#endif
}

__device__ __forceinline__ void wait_async_zero() {
#if defined(__gfx1250__) && __has_builtin(__builtin_amdgcn_s_wait_asynccnt)
    __builtin_amdgcn_s_wait_asynccnt(0);
#else
    asm volatile("s_wait_asynccnt 0" ::: "memory");
#endif
}

// ---- kernel ---------------------------------------------------------------
// One workgroup (256 threads / 8 waves) per ROI, double-buffered async
// staging of the 16 bilinear corner vectors per pooled cell.
__global__ __launch_bounds__(256)
void RoipoolingLayer_56203942035848_kernel(const float* __restrict__ feat,
                                           const float* __restrict__ rois,
                                           const int*   __restrict__ im_size,
                                           float*       __restrict__ out) {
    __shared__ int   sx0[SAMP], sx1[SAMP], sy0[SAMP], sy1[SAMP];
    __shared__ float swx[SAMP], swy[SAMP];
    __shared__ __align__(16) float corners[2][16][CCH];   // 2 x 32 KB staging

    const int n   = blockIdx.x;
    const int tid = threadIdx.x;
    const float* roi = rois + 4 * (size_t)n;

    // ---- per-ROI sample grid metadata (TF crop_and_resize convention) ----
    if (tid < SAMP) {
        const float w   = (float)im_size[1];
        const float t   = (float)tid * (1.0f / (float)(SAMP - 1));
        const float nx1 = roi[0] / w;
        const float nx2 = roi[2] / w;
        float xs = (nx1 + (nx2 - nx1) * t) * (float)(WFEAT - 1);
        xs = fminf(fmaxf(xs, 0.0f), (float)(WFEAT - 1));
        const float x0f = floorf(xs);
        const int   x0  = (int)x0f;
        sx0[tid] = x0;
        sx1[tid] = min(x0 + 1, WFEAT - 1);
        swx[tid] = xs - x0f;
    } else if (tid >= 32 && tid < 32 + SAMP) {
        const int   s   = tid - 32;
        const float h   = (float)im_size[0];
        const float t   = (float)s * (1.0f / (float)(SAMP - 1));
        const float ny1 = roi[1] / h;
        const float ny2 = roi[3] / h;
        float ys = (ny1 + (ny2 - ny1) * t) * (float)(HFEAT - 1);
        ys = fminf(fmaxf(ys, 0.0f), (float)(HFEAT - 1));
        const float y0f = floorf(ys);
        const int   y0  = (int)y0f;
        sy0[s] = y0;
        sy1[s] = min(y0 + 1, HFEAT - 1);
        swy[s] = ys - y0f;
    }
    __syncthreads();

    const int ch4 = (tid & 127) * 4;   // 128 lanes x float4 cover one 512-ch corner
    const int c2  = tid * 2;           // blend phase: 2 channels per thread

    // Issue the 8 async b128 loads that stage one cell's 16 corner vectors.
    // Corner index k = ry*4 + rx; ry = 2*i + a (sample-row i, y0/y1 corner a),
    // rx = 2*j + b.
    auto stage_cell = [&](int cell, float (*dstbuf)[CCH]) {
        const int py  = cell / POOLD;
        const int px  = cell - POOLD * py;
        const int syA = 2 * py;
        const int sxA = 2 * px;
        #pragma unroll
        for (int kk = (tid >> 7); kk < 16; kk += 2) {
            const int ry = kk >> 2;
            const int rx = kk & 3;
            const int fy = (ry & 1) ? sy1[syA + (ry >> 1)] : sy0[syA + (ry >> 1)];
            const int fx = (rx & 1) ? sx1[sxA + (rx >> 1)] : sx0[sxA + (rx >> 1)];
            const float* src = feat + ((size_t)(fy * WFEAT + fx)) * CCH + ch4;
            async_copy_b128_to_lds(src, &dstbuf[kk][ch4]);
        }
    };

    // Prologue: stage cell 0 into buffer 0.
    stage_cell(0, corners[0]);

    for (int cell = 0; cell < NCELL; ++cell) {
        const int buf = cell & 1;

        // Prefetch next cell into the other buffer, then drain this cell's
        // 8 loads (in-order completion => ASYNCcnt<=8 suffices).
        if (cell + 1 < NCELL) {
            stage_cell(cell + 1, corners[buf ^ 1]);
            wait_async_le8();
        } else {
            wait_async_zero();
        }
        __syncthreads();       // all waves' staging for `cell` visible in LDS

        const int py  = cell / POOLD;
        const int px  = cell - POOLD * py;
        const int syA = 2 * py;
        const int sxA = 2 * px;

        // ---- blend 4 samples, max-pool, store ----
        f2 acc;
        acc.x = -3.402823466e+38f;
        acc.y = -3.402823466e+38f;
        #pragma unroll
        for (int i = 0; i < 2; ++i) {
            const float wy  = swy[syA + i];
            const float wy0 = 1.0f - wy;
            #pragma unroll
            for (int j = 0; j < 2; ++j) {
                const float wx  = swx[sxA + j];
                const float wx0 = 1.0f - wx;
                const f2 c00 = *(const f2*)&corners[buf][(2 * i + 0) * 4 + (2 * j + 0)][c2];
                const f2 c01 = *(const f2*)&corners[buf][(2 * i + 0) * 4 + (2 * j + 1)][c2];
                const f2 c10 = *(const f2*)&corners[buf][(2 * i + 1) * 4 + (2 * j + 0)][c2];
                const f2 c11 = *(const f2*)&corners[buf][(2 * i + 1) * 4 + (2 * j + 1)][c2];
                const float w00 = wy0 * wx0, w01 = wy0 * wx, w10 = wy * wx0, w11 = wy * wx;
                const float vx = c00.x * w00 + c01.x * w01 + c10.x * w10 + c11.x * w11;
                const float vy = c00.y * w00 + c01.y * w01 + c10.y * w10 + c11.y * w11;
                acc.x = fmaxf(acc.x, vx);
                acc.y = fmaxf(acc.y, vy);
            }
        }
        // Streaming output (written once, never re-read): non-temporal b64.
        __builtin_nontemporal_store(acc, (f2*)&out[((size_t)n * NCELL + cell) * CCH + c2]);

        __syncthreads();       // readers done before this parity buffer is re-staged
    }
}

extern "C" void kernel_launch(void* const* d_in, const int* in_sizes, int n_in,
                              void* d_out, int out_size, void* d_ws, size_t ws_size,
                              hipStream_t stream) {
    (void)n_in; (void)out_size; (void)d_ws; (void)ws_size;
    const float* feat    = (const float*)d_in[0];   // [1,37,50,512] f32
    const float* rois    = (const float*)d_in[1];   // [N,4] f32
    const int*   im_size = (const int*)d_in[2];     // [2] i32
    float*       out     = (float*)d_out;           // [N,7,7,512] f32
    const int N = in_sizes[1] / 4;
    if (N <= 0) return;
    RoipoolingLayer_56203942035848_kernel<<<N, 256, 0, stream>>>(feat, rois, im_size, out);
}